// Affinity_Propagate_58806692216891
// MI455X (gfx1250) — compile-verified
//
#include <hip/hip_runtime.h>

// Volume geometry (fixed by the reference): guidance [24,256,256,32], blur [1,256,256,32]
#define DX_ 256
#define DY_ 256
#define DZ_ 32
#define SX_ (DY_*DZ_)      // 8192 elements: x stride
#define SY_ (DZ_)          // 32 elements:   y stride
#define CH_ (DX_*DY_*DZ_)  // 2097152:       guidance channel stride

// neighbor offset tables: d = (1 - top_pad, 1 - left_pad) for the 8 pads
static constexpr int DH_[8] = { 1, 1, 1, 0, 0,-1,-1,-1};
static constexpr int DW_[8] = { 1, 0,-1, 1,-1, 1, 0,-1};

#if __has_builtin(__builtin_amdgcn_global_load_async_to_lds_b32)
#define AP_ASYNC_LDS 1
#else
#define AP_ASYNC_LDS 0
#endif

typedef __attribute__((address_space(1))) int ap_gint_t;
typedef __attribute__((address_space(3))) int ap_lint_t;

__device__ __forceinline__ void ap_copy_f32(const float* gsrc, float* ldst) {
#if AP_ASYNC_LDS
    __builtin_amdgcn_global_load_async_to_lds_b32(
        (ap_gint_t*)gsrc, (ap_lint_t*)ldst, /*offset=*/0, /*cpol=*/0);
#else
    *ldst = *gsrc;
#endif
}

__device__ __forceinline__ void ap_async_wait() {
#if AP_ASYNC_LDS
#if __has_builtin(__builtin_amdgcn_s_wait_asynccnt)
    __builtin_amdgcn_s_wait_asynccnt(0);
#else
    asm volatile("s_wait_asynccnt 0" ::: "memory");
#endif
#endif
}

// MODE 0: x-view  — stencil in (x,y), slab z. guidance ch 0..7,  g[k, x+dH, y+dW, z]
// MODE 1: y-view  — stencil in (x,z), slab y. guidance ch 8..15, g[8+k, x+dH, y, z+dW]
// MODE 2: z-view  — stencil in (y,z), slab x. guidance ch 16..23,g[16+k, x, y+dH, z+dW]
template <int MODE>
__global__ __launch_bounds__(256)
void ap_prop_step(const float* __restrict__ g,
                  const float* __restrict__ src,
                  float* __restrict__ dst)
{
    const int tid = threadIdx.x;
    const int tz  = tid & 31;   // z lane, contiguous in memory
    const int tr  = tid >> 5;   // row within block, 0..7

    if constexpr (MODE == 0) {
        // tile: 3 x-planes x 10 y-rows x 32 z
        __shared__ float tile[3 * 10 * 32];
        const int x  = blockIdx.y;
        const int y0 = blockIdx.x * 8;
        const int y  = y0 + tr;
        const int z  = tz;

        for (int i = tid; i < 3 * 10 * 32; i += 256) tile[i] = 0.0f;
        __syncthreads();

        for (int i = tid; i < 3 * 10 * 32; i += 256) {
            const int pp  = i / 320;       // x-plane 0..2
            const int rem = i - pp * 320;
            const int rr  = rem >> 5;      // y-row 0..9
            const int zz  = rem & 31;
            const int xc  = x - 1 + pp;
            const int yc  = y0 - 1 + rr;
            if ((unsigned)xc < DX_ && (unsigned)yc < DY_)
                ap_copy_f32(src + (size_t)xc * SX_ + yc * SY_ + zz, &tile[i]);
        }
        ap_async_wait();
        __syncthreads();

        const float rc = tile[1 * 320 + (tr + 1) * 32 + tz];
        float A = 0.0f, S = 0.0f, W = 0.0f;
#pragma unroll
        for (int k = 0; k < 8; ++k) {
            const int dx = DH_[k], dy = DW_[k];
            const int xn = x + dx, yn = y + dy;
            float gv = 0.0f;
            if ((unsigned)xn < DX_ && (unsigned)yn < DY_)
                gv = g[(size_t)k * CH_ + (size_t)xn * SX_ + yn * SY_ + z];
            const float rn = tile[(1 + dx) * 320 + (tr + 1 + dy) * 32 + tz];
            A += fabsf(gv);
            S += gv;
            W = __builtin_fmaf(gv, rn, W);
        }
        dst[(size_t)x * SX_ + y * SY_ + z] = rc + (W - S * rc) / A;
    } else {
        // tile: 10 R-rows x (32+2) z, zeroed z halo columns
        __shared__ float tile[10 * 34];
        const int p  = blockIdx.y;        // fixed plane coord (y for MODE1, x for MODE2)
        const int r0 = blockIdx.x * 8;    // R = x (MODE1) or y (MODE2)
        const int rC = r0 + tr;
        const int z  = tz;
        const int x  = (MODE == 1) ? rC : p;
        const int y  = (MODE == 1) ? p  : rC;

        for (int i = tid; i < 10 * 34; i += 256) tile[i] = 0.0f;
        __syncthreads();

        for (int rr = tr; rr < 10; rr += 8) {
            const int rcoord = r0 - 1 + rr;
            if ((unsigned)rcoord < 256u) {
                const int xc = (MODE == 1) ? rcoord : p;
                const int yc = (MODE == 1) ? p : rcoord;
                ap_copy_f32(src + (size_t)xc * SX_ + yc * SY_ + tz,
                            &tile[rr * 34 + 1 + tz]);
            }
        }
        ap_async_wait();
        __syncthreads();

        const float rc = tile[(tr + 1) * 34 + tz + 1];
        float A = 0.0f, S = 0.0f, W = 0.0f;
        constexpr int chb = (MODE == 1) ? 8 : 16;
#pragma unroll
        for (int k = 0; k < 8; ++k) {
            const int dR = DH_[k], dz = DW_[k];
            const int Rn = rC + dR, zn = z + dz;
            float gv = 0.0f;
            if ((unsigned)Rn < 256u && (unsigned)zn < DZ_) {
                const int xn = (MODE == 1) ? Rn : p;
                const int yn = (MODE == 1) ? p : Rn;
                gv = g[(size_t)(chb + k) * CH_ + (size_t)xn * SX_ + yn * SY_ + zn];
            }
            const float rn = tile[(tr + 1 + dR) * 34 + tz + 1 + dz];
            A += fabsf(gv);
            S += gv;
            W = __builtin_fmaf(gv, rn, W);
        }
        dst[(size_t)x * SX_ + y * SY_ + z] = rc + (W - S * rc) / A;
    }
}

extern "C" void kernel_launch(void* const* d_in, const int* in_sizes, int n_in,
                              void* d_out, int out_size, void* d_ws, size_t ws_size,
                              hipStream_t stream) {
    (void)in_sizes; (void)n_in; (void)out_size; (void)ws_size;
    const float* g    = (const float*)d_in[0];  // guidance [24,256,256,32]
    const float* blur = (const float*)d_in[1];  // blur     [1,256,256,32]
    float* out = (float*)d_out;
    float* w0  = (float*)d_ws;                  // needs 2M floats = 8.4 MB

    const dim3 grid(32, 256), block(256);
    const float* src = blur;
    for (int step = 0; step < 9; ++step) {
        float* dst = ((step & 1) == 0) ? out : w0;   // step 8 (last) -> out
        switch (step % 3) {
        case 0: ap_prop_step<0><<<grid, block, 0, stream>>>(g, src, dst); break;
        case 1: ap_prop_step<1><<<grid, block, 0, stream>>>(g, src, dst); break;
        case 2: ap_prop_step<2><<<grid, block, 0, stream>>>(g, src, dst); break;
        }
        src = dst;
    }
}